// Linear_74534862454835
// MI455X (gfx1250) — compile-verified
//
#include <hip/hip_runtime.h>
#include <cstdint>

#define FEAT     136
#define CHUNKS   34            // 136 floats = 34 x 16B chunks, exact
#define ROWPITCH 140           // padded row pitch in dwords (560B): conflict-minimal b128 reads
#define ROWS     112           // rows per block tile -> 112*560B = 62.7 KB LDS
#define THREADS  128           // 4 wave32; one row per thread in compute phase
#define LN_EPS   1e-5f

typedef __attribute__((ext_vector_type(2))) float v2f;
typedef __attribute__((ext_vector_type(8))) float v8f;

// CDNA5 async global->LDS copy, 16B per lane, ASYNCcnt-tracked, non-temporal
// (x is streamed exactly once; don't pollute L2).
__device__ __forceinline__ void async_load16(uint32_t lds_off, uint64_t gaddr) {
    asm volatile("global_load_async_to_lds_b128 %0, %1, off th:TH_LOAD_NT"
                 :: "v"(lds_off), "v"(gaddr)
                 : "memory");
}
__device__ __forceinline__ void wait_async0() {
    asm volatile("s_wait_asynccnt 0" ::: "memory");
}

__global__ __launch_bounds__(THREADS)
void ln_linear_kernel(const float* __restrict__ x,
                      const float* __restrict__ gamma,
                      const float* __restrict__ beta,
                      const float* __restrict__ w,
                      const float* __restrict__ b,
                      float* __restrict__ out,
                      int n_rows)
{
    __shared__ __align__(16) float tile[ROWS * ROWPITCH];   // padded rows
    __shared__ __align__(16) float sgw[FEAT];               // gamma*w
    __shared__ float sGC[2];                                // {sum(gamma*w), sum(beta*w)+b}

    const int tid  = threadIdx.x;
    const long long row0 = (long long)blockIdx.x * ROWS;
    const long long rem  = n_rows - row0;
    const int rows = (int)(rem < ROWS ? rem : ROWS);
    if (rows <= 0) return;

    // ---- 1) async DMA of the x tile: global side contiguous (coalesced),
    //         LDS side scattered into the 560B-pitch padded layout ----
    {
        const int chunks = rows * CHUNKS;                               // 16B chunks
        const uint64_t gbase = (uint64_t)(uintptr_t)(x + row0 * FEAT);  // 544B rows, 16B aligned
        const uint32_t lbase = (uint32_t)(uintptr_t)(&tile[0]);         // low 32 bits = LDS offset
        for (int c = tid; c < chunks; c += THREADS) {
            const unsigned r = (unsigned)c / (unsigned)CHUNKS;          // magic-multiply div
            const unsigned m = (unsigned)c - r * CHUNKS;
            async_load16(lbase + r * (ROWPITCH * 4u) + m * 16u,
                         gbase + (uint64_t)c * 16u);
        }
    }

    // ---- 2a) fold gamma*w into LDS while the DMA is in flight ----
    for (int j = tid; j < FEAT; j += THREADS) sgw[j] = gamma[j] * w[j];

    // ---- 2b) G = gamma.w and Cw = beta.w as a WMMA dot-product (wave 0) ----
    // A (16x4 f32): row 0 = gamma chunk, row 1 = beta chunk, rows 2..15 = 0.
    //   Layout: lanes 0-15 = M, VGPR0/1 = K0/K1; lanes 16-31 = same M, K2/K3.
    // B (4x16 f32): column 0 = w chunk, columns 1..15 = 0.
    //   Layout: lanes 0-15 = N with K0 (V0) / K1 (V1); lanes 16-31 = K2/K3.
    // D (16x16 f32, 8 VGPRs): D[0][0] -> acc[0]@lane0, D[1][0] -> acc[1]@lane0.
    if (tid < 32) {                         // whole wave 0: EXEC all ones (WMMA requirement)
        const int lane  = tid;
        const int m16   = lane & 15;
        const int khalf = lane >> 4;        // 0 -> K0,K1 ; 1 -> K2,K3
        v8f acc = {};
        for (int k = 0; k < CHUNKS; ++k) {
            const int kb = 4 * k + 2 * khalf;               // even -> 8B aligned
            const v2f g2 = *(const v2f*)(gamma + kb);
            const v2f b2 = *(const v2f*)(beta  + kb);
            const v2f w2 = *(const v2f*)(w     + kb);
            const v2f zz = {0.0f, 0.0f};
            const v2f a  = (m16 == 0) ? g2 : ((m16 == 1) ? b2 : zz);
            const v2f bb = (m16 == 0) ? w2 : zz;            // N==0 column only
            acc = __builtin_amdgcn_wmma_f32_16x16x4_f32(
                      false, a, false, bb, (short)0, acc, false, false);
        }
        if (lane == 0) {
            sGC[0] = acc[0];            // G  = sum gamma*w   (D[0][0])
            sGC[1] = acc[1] + b[0];     // C  = sum beta*w + b (D[1][0])
        }
    }
    __syncthreads();                       // sgw / sGC visible

    const float G = sGC[0];
    const float C = sGC[1];

    wait_async0();                         // drain this wave's ASYNCcnt
    __syncthreads();                       // tile fully resident for all waves

    // ---- 3) one THREAD per row: no cross-lane traffic, coalesced store ----
    if (tid < rows) {
        const float* rowp = tile + tid * ROWPITCH;
        float s1a = 0.0f, s1b = 0.0f;                  // sum x
        float s2a = 0.0f, s2b = 0.0f;                  // sum x^2
        float s3a = 0.0f, s3b = 0.0f;                  // sum x*(gamma*w)
        #pragma unroll
        for (int j = 0; j < CHUNKS; ++j) {
            const float4 xv = *(const float4*)(rowp + j * 4);   // conflict-minimal (pitch 140 dw)
            const float4 gv = *(const float4*)(sgw  + j * 4);   // broadcast (same addr all lanes)
            s1a += xv.x + xv.z;
            s1b += xv.y + xv.w;
            s2a = fmaf(xv.x, xv.x, fmaf(xv.z, xv.z, s2a));
            s2b = fmaf(xv.y, xv.y, fmaf(xv.w, xv.w, s2b));
            s3a = fmaf(xv.x, gv.x, fmaf(xv.z, gv.z, s3a));
            s3b = fmaf(xv.y, gv.y, fmaf(xv.w, gv.w, s3b));
        }
        const float inv  = 1.0f / (float)FEAT;
        const float mean = (s1a + s1b) * inv;
        const float var  = fmaf(-mean, mean, (s2a + s2b) * inv);   // E[x^2] - mean^2
        const float ve   = var + LN_EPS;
        float rstd = rsqrtf(ve);
        rstd = rstd * fmaf(-0.5f * ve, rstd * rstd, 1.5f);         // one Newton step
        out[row0 + tid] = fmaf(rstd, fmaf(-mean, G, s3a + s3b), C);
    }
}

extern "C" void kernel_launch(void* const* d_in, const int* in_sizes, int n_in,
                              void* d_out, int out_size, void* d_ws, size_t ws_size,
                              hipStream_t stream) {
    const float* x     = (const float*)d_in[0];
    const float* gamma = (const float*)d_in[1];
    const float* beta  = (const float*)d_in[2];
    const float* w     = (const float*)d_in[3];
    const float* b     = (const float*)d_in[4];
    float* out = (float*)d_out;

    const int n_rows = in_sizes[0] / FEAT;
    const int grid   = (n_rows + ROWS - 1) / ROWS;
    ln_linear_kernel<<<grid, THREADS, 0, stream>>>(x, gamma, beta, w, b, out, n_rows);
}